// Or_Convolution_2d_56564719288840
// MI455X (gfx1250) — compile-verified
//
#include <hip/hip_runtime.h>
#include <hip/hip_bf16.h>
#include <stdint.h>

// Problem constants (from the reference)
#define N_ROWS 4096
#define L_COLS 4096
#define CH     8
#define KERN   16
#define STEP   2              // STRIDE + 1
#define W_OUT  2041           // (4096 - 16)/2 + 1
#define WPB    256            // windows per block (one per thread)
#define COLS_PB (STEP*(WPB-1) + KERN)   // 526 temp columns needed per block

__global__ __launch_bounds__(256)
void or_conv2d_kernel(const float* __restrict__ x,
                      const float* __restrict__ wt,
                      float* __restrict__ out)
{
    // x tile: 526 cols * 8 ch * 4B = 16832 B ; temp: 526*4 = 2104 B (+pad)
    __shared__ __align__(16) float xs[COLS_PB * CH];
    __shared__ __align__(16) float ts[COLS_PB + 2];

    const int tid = threadIdx.x;
    const int row = blockIdx.y;
    const int w0  = blockIdx.x * WPB;
    const int c0  = w0 * STEP;

    // ---------- Phase 1: async DMA of the x tile into LDS (gfx1250 path) ----------
    const uint32_t xs_base = (uint32_t)(uintptr_t)&xs[0];   // low 32 bits of flat = LDS offset
    for (int c = tid; c < COLS_PB; c += 256) {
        int gcol = c0 + c;
        if (gcol > L_COLS - 1) gcol = L_COLS - 1;           // clamp: values never consumed
        uint64_t gaddr = (uint64_t)(uintptr_t)(x + ((size_t)row * L_COLS + (size_t)gcol) * CH);
        uint32_t laddr = xs_base + (uint32_t)c * (CH * 4);
        // 32 B per column = two b128 async transfers; INST_OFFSET applies to
        // both the global and the LDS address, so reuse the same regs.
        asm volatile("global_load_async_to_lds_b128 %0, %1, off"
                     :: "v"(laddr), "v"(gaddr) : "memory");
        asm volatile("global_load_async_to_lds_b128 %0, %1, off offset:16"
                     :: "v"(laddr), "v"(gaddr) : "memory");
    }
    asm volatile("s_wait_asynccnt 0x0" ::: "memory");
    __syncthreads();

    // ---------- Phase 2: channel max (8 -> 1) into temp LDS ----------
    for (int c = tid; c < COLS_PB; c += 256) {
        const float4* p = (const float4*)&xs[c * CH];
        float4 a = p[0];
        float4 b = p[1];
        float m0 = fmaxf(fmaxf(a.x, a.y), fmaxf(a.z, a.w));
        float m1 = fmaxf(fmaxf(b.x, b.y), fmaxf(b.z, b.w));
        ts[c] = fmaxf(m0, m1);
    }
    __syncthreads();

    // ---------- Phase 3: weighted windowed max ----------
    const int w = w0 + tid;
    if (w < W_OUT) {
        float wk[KERN];
        #pragma unroll
        for (int k = 0; k < KERN; ++k) wk[k] = wt[k];

        const float2* t2 = (const float2*)&ts[tid * STEP];  // 8-byte aligned
        float m = -__builtin_inff();
        #pragma unroll
        for (int j = 0; j < KERN / 2; ++j) {
            float2 v = t2[j];
            m = fmaxf(m, fmaxf(v.x * wk[2*j], v.y * wk[2*j + 1]));
        }
        out[(size_t)row * W_OUT + w] = m;
    }
}

extern "C" void kernel_launch(void* const* d_in, const int* in_sizes, int n_in,
                              void* d_out, int out_size, void* d_ws, size_t ws_size,
                              hipStream_t stream) {
    const float* x  = (const float*)d_in[0];   // (4096, 4096, 8) f32
    const float* wt = (const float*)d_in[1];   // (1, 16) f32
    float* out = (float*)d_out;                // (4096, 2041) f32

    dim3 grid((W_OUT + WPB - 1) / WPB, N_ROWS);  // (8, 4096)
    or_conv2d_kernel<<<grid, 256, 0, stream>>>(x, wt, out);
}